// approxconv2d_65841848648390
// MI455X (gfx1250) — compile-verified
//
#include <hip/hip_runtime.h>

// ApproxConv2d via 256x256 LUT gathers, LUT resident in CDNA5's 320KB LDS,
// filled with the gfx1250 async global->LDS DMA path (ASYNCcnt).
//
// Shapes (fixed by setup_inputs):
//   x      : [8, 64, 32, 32] f32
//   weight : [64, 64, 3, 3]  f32
//   bias   : [64]            f32
//   lut    : [256, 256]      f32
//   out    : [8, 64, 32, 32] f32

#define B_   8
#define C_   64
#define H_   32
#define W_   32
#define O_   64
#define KK   3
#define P_   (C_ * KK * KK)     // 576
#define OH_  32
#define OW_  32

#define LUT_BYTES  (256 * 256 * 4)          // 262144
#define QWT_OFF    LUT_BYTES                // 262144, 8B-aligned
#define QWT_BYTES  (P_ * O_)                // 36864
#define QX_OFF     (QWT_OFF + QWT_BYTES)    // 299008
#define QX_BYTES   (P_ * OW_)               // 18432
#define SMEM_BYTES (QX_OFF + QX_BYTES)      // 317440 < 320KB WGP LDS

__device__ __forceinline__ int quant_idx(float v) {
    int qi = (int)rintf(v * 64.0f);          // RNE == jnp.round
    qi = qi < -128 ? -128 : (qi > 127 ? 127 : qi);
    return qi + 128;                         // [0, 255]
}

__global__ __launch_bounds__(256)
void approxconv2d_65841848648390_kernel(const float* __restrict__ x,
                                        const float* __restrict__ weight,
                                        const float* __restrict__ bias,
                                        const float* __restrict__ lut,
                                        float* __restrict__ out) {
    extern __shared__ __align__(16) unsigned char smem[];
    float*         lutf = (float*)smem;                 // [65536] f32
    unsigned char* qwT  = smem + QWT_OFF;               // [P_][O_] u8
    unsigned char* qxs  = smem + QX_OFF;                // [P_][OW_] u8

    const int t = threadIdx.x;
    const int b = blockIdx.x >> 5;                      // 0..7
    const int h = blockIdx.x & 31;                      // 0..31

    // ---------- Phase 1: issue async DMA of the 256KB LUT into LDS ----------
    // 16384 x 16B lane-transfers; each wave32 instruction moves 512B.
    {
        unsigned           lds0 = (unsigned)(unsigned long long)(void*)lutf;
        unsigned long long g0   = (unsigned long long)lut;
#pragma unroll 4
        for (int i = 0; i < 64; ++i) {
            unsigned           off    = (unsigned)((i * 256 + t) * 16);
            unsigned           ldsdst = lds0 + off;
            unsigned long long gaddr  = g0 + off;
            asm volatile("global_load_async_to_lds_b128 %0, %1, off"
                         :: "v"(ldsdst), "v"(gaddr)
                         : "memory");
        }
    }

    // ---------- Phase 2 (overlapped with DMA): quantize weights ----------
    // weight is already [O][P] with P = (c,kh,kw) channel-slowest, matching
    // the reference's reshape(O, P). Store transposed: qwT[p*64 + o].
    for (int idx = t; idx < O_ * P_; idx += 256) {
        int o = idx / P_;
        int p = idx - o * P_;
        qwT[p * O_ + o] = (unsigned char)quant_idx(weight[idx]);
    }

    // ---------- Phase 3 (overlapped with DMA): quantize this (b,h) column ---
    // Zero padding maps to index 128; lut row 128 is exactly zero
    // (0 * (1+eps) == 0), matching the reference bit-for-bit.
    for (int idx = t; idx < P_ * OW_; idx += 256) {
        int p  = idx >> 5;
        int w  = idx & 31;
        int c  = p / 9;
        int r  = p - c * 9;
        int kh = r / 3;
        int kw = r - kh * 3;
        int ih = h + kh - 1;
        int iw = w + kw - 1;
        float v = 0.0f;
        if ((unsigned)ih < 32u && (unsigned)iw < 32u)
            v = x[(((b * C_) + c) * H_ + ih) * W_ + iw];
        qxs[idx] = (unsigned char)quant_idx(v);
    }

    // Fence the async LUT fill, then sync all 8 waves.
    asm volatile("s_wait_asynccnt 0" ::: "memory");
    __syncthreads();

    // ---------- Phase 4: LDS-gather accumulation ----------
    // Thread t: output column w = t&31, 8 contiguous channels o = o8*8..o8*8+7.
    const int w  = t & 31;
    const int o8 = t >> 5;                              // 0..7
    const unsigned long long* qwT64 =
        (const unsigned long long*)(smem + QWT_OFF);    // 8B-aligned rows

    float acc[8];
#pragma unroll
    for (int u = 0; u < 8; ++u) acc[u] = 0.0f;

#pragma unroll 2
    for (int p = 0; p < P_; ++p) {
        int base = ((int)qxs[p * OW_ + w]) << 8;        // ds_load_u8
        unsigned long long q8 = qwT64[p * 8 + o8];      // ds_load_b64: 8 qw idx
        unsigned lo = (unsigned)q8;
        unsigned hi = (unsigned)(q8 >> 32);
        acc[0] += lutf[base + (lo & 255)];              // 8x ds_load_b32 gather
        acc[1] += lutf[base + ((lo >> 8) & 255)];
        acc[2] += lutf[base + ((lo >> 16) & 255)];
        acc[3] += lutf[base + (lo >> 24)];
        acc[4] += lutf[base + (hi & 255)];
        acc[5] += lutf[base + ((hi >> 8) & 255)];
        acc[6] += lutf[base + ((hi >> 16) & 255)];
        acc[7] += lutf[base + (hi >> 24)];
    }

#pragma unroll
    for (int u = 0; u < 8; ++u) {
        int o = o8 * 8 + u;
        out[(((b * O_) + o) * OH_ + h) * OW_ + w] = acc[u] + bias[o];
    }
}

extern "C" void kernel_launch(void* const* d_in, const int* in_sizes, int n_in,
                              void* d_out, int out_size, void* d_ws, size_t ws_size,
                              hipStream_t stream) {
    const float* x      = (const float*)d_in[0];
    const float* weight = (const float*)d_in[1];
    const float* bias   = (const float*)d_in[2];
    const float* lut    = (const float*)d_in[3];
    float*       out    = (float*)d_out;

    dim3 grid(B_ * H_);   // 256 blocks = (b, h)
    dim3 block(256);      // 8 wave32s
    approxconv2d_65841848648390_kernel<<<grid, block, SMEM_BYTES, stream>>>(
        x, weight, bias, lut, out);
}